// WaveletModel_26293789786554
// MI455X (gfx1250) — compile-verified
//
#include <hip/hip_runtime.h>
#include <math.h>

// CDNA5 / gfx1250 CWT kernel.
//  * Morlet envelope anchored at t=0 dies below f32 eps by t=512  -> 512-tap direct conv
//  * output channel i uses (c=i%4, s=i%32) and i%4==(i%32)%4      -> 4x duplicated channels
//  * x is real -> complex conv = two real convs -> GEMM [16384x512]x[512x16] per (b,c)
//  * f32 WMMA 16x16x4 keeps exact f32 accumulation (matches reference precision)

typedef __attribute__((ext_vector_type(2))) float v2f;
typedef __attribute__((ext_vector_type(8))) float v8f;

#define NSCALES 32
#define NCH     128        // C*S output channels
#define NPIX    16384      // H*W
#define TAPS    512        // effective wavelet support (envelope < 6e-16 beyond)
#define SHIFT   8191       // y[n] = sum_t x[n-8191+t] * wav[t]
#define CHS     (NPIX * 2) // floats per output channel (interleaved complex64)

// ---------------------------------------------------------------------------
// Init kernel: wavelet bank wbank[c][k][m], m = 16 rows = 8 scales x {re,im}.
// Row m -> global scale s = c + 4*(m>>1), component = m&1 (0=cos,1=sin).
// ---------------------------------------------------------------------------
__global__ __launch_bounds__(256) void cwt_wbank_init(const float* __restrict__ scales,
                                                      float* __restrict__ wbank) {
    int gid = blockIdx.x * 256 + threadIdx.x;       // [0, 4*512*16)
    if (gid >= 4 * TAPS * 16) return;
    int m = gid & 15;
    int k = (gid >> 4) & (TAPS - 1);
    int c = gid >> 13;                              // 512*16 = 2^13
    int s = c + 4 * (m >> 1);
    float sc = scales[s];
    float fr = 1.0f / sc;
    float sigma = sc * (6.0f / (2.0f * 3.14159265358979323846f));
    float env = expf(-((float)k * (float)k) / (2.0f * sigma * sigma));
    double ph = 2.0 * 3.141592653589793 * (double)fr * (double)k;
    float trig = (m & 1) ? (float)sin(ph) : (float)cos(ph);
    wbank[gid] = env * trig;
}

// ---------------------------------------------------------------------------
// Main kernel: grid = (NPIX/256, 16 (b,c) pairs), block = 256 threads (8 waves).
// Each wave: two 16x16 D tiles (32 consecutive n), all 16 scale-components.
// A = wavelet (LDS), B = Hankel x window (LDS), D = f32 accum via v_wmma.
// ---------------------------------------------------------------------------
__global__ __launch_bounds__(256) void cwt_wmma_kernel(const float* __restrict__ x,
                                                       const float* __restrict__ wbank,
                                                       float* __restrict__ out) {
    __shared__ float xwin[256 + TAPS];      // 768 floats: x window with zero padding
    __shared__ float wlds[TAPS * 16];       // 32 KB: wavelet slice for this c

    const int tid    = threadIdx.x;
    const int bc     = blockIdx.y;
    const int b      = bc >> 2;
    const int c      = bc & 3;
    const int n0_blk = blockIdx.x * 256;

    // stage x window (with linear-conv zero padding) into LDS
    const float* __restrict__ xc = x + (size_t)(b * 4 + c) * NPIX;
    for (int i = tid; i < 256 + TAPS; i += 256) {
        int idx = n0_blk - SHIFT + i;
        float v = 0.0f;
        if (idx >= 0 && idx < NPIX) v = xc[idx];
        xwin[i] = v;
    }
    // stage wavelet bank slice (L2-resident, read once per block)
    const float* __restrict__ wsrc = wbank + (size_t)c * (TAPS * 16);
    for (int i = tid; i < TAPS * 16; i += 256) wlds[i] = wsrc[i];
    __syncthreads();

    const int w  = tid >> 5;     // wave id 0..7
    const int l  = tid & 31;     // lane
    const int jn = l & 15;       // N-col within tile / A row
    const int hi = l >> 4;       // lane half -> K pair select

    v8f acc[2] = {};

    // strength-reduced LDS pointers: wp walks the wavelet rows, xp the Hankel window
    const float* wp = &wlds[2 * hi * 16 + jn];
    const float* xp = &xwin[w * 32 + jn + 2 * hi];

#pragma unroll 2
    for (int k0 = 0; k0 < TAPS; k0 += 4) {
        // A fragment: 16x4 wavelet tile. lane: row=jn, VGPR0=K(k), VGPR1=K(k+1)
        v2f a;
        a.x = wp[0];
        a.y = wp[16];
        // B fragments: 4x16 Hankel x tiles (two n-tiles share A)
        v2f bf0, bf1;
        bf0.x = xp[0];
        bf0.y = xp[1];
        bf1.x = xp[16];
        bf1.y = xp[17];
        acc[0] = __builtin_amdgcn_wmma_f32_16x16x4_f32(
            false, a, false, bf0, (short)0, acc[0], false, false);
        acc[1] = __builtin_amdgcn_wmma_f32_16x16x4_f32(
            false, a, false, bf1, (short)0, acc[1], false, false);
        wp += 64;   // next K-quad of wavelet rows (4 * 16 floats)
        xp += 4;    // next K-quad of x window
    }

    // Store: D VGPR pair (2r, 2r+1) = (re, im) of one scale, lane%16 = n offset.
    // Coalesced global_store_b64 of interleaved complex64, duplicated 4x (CRT).
    // Constant channel strides fold into the 24-bit instruction offset; two base
    // pointers (q in {0,1} / q in {2,3}) keep every offset under 8 MB.
    float* const pbase = out + ((size_t)b * NCH + c + 16 * hi) * CHS;
#pragma unroll
    for (int t = 0; t < 2; ++t) {
        const int n = n0_blk + w * 32 + t * 16 + jn;
        float* const p0 = pbase + (size_t)n * 2;            // q = 0,1 copies
        float* const p1 = p0 + (size_t)64 * CHS;            // q = 2,3 copies
#pragma unroll
        for (int rp = 0; rp < 4; ++rp) {
            float2 v;
            v.x = acc[t][2 * rp];        // re
            v.y = acc[t][2 * rp + 1];    // im
            *(float2*)(p0 + (4 * rp     ) * CHS) = v;       // q=0
            *(float2*)(p0 + (4 * rp + 32) * CHS) = v;       // q=1
            *(float2*)(p1 + (4 * rp     ) * CHS) = v;       // q=2
            *(float2*)(p1 + (4 * rp + 32) * CHS) = v;       // q=3
        }
    }
}

// ---------------------------------------------------------------------------
extern "C" void kernel_launch(void* const* d_in, const int* in_sizes, int n_in,
                              void* d_out, int out_size, void* d_ws, size_t ws_size,
                              hipStream_t stream) {
    (void)in_sizes; (void)n_in; (void)out_size; (void)ws_size;
    const float* x      = (const float*)d_in[0];   // [4,4,128,128] f32
    const float* scales = (const float*)d_in[1];   // [32] f32
    float* out   = (float*)d_out;                  // [4,128,16384] complex64 interleaved
    float* wbank = (float*)d_ws;                   // 4*512*16 floats = 128 KB scratch

    cwt_wbank_init<<<dim3((4 * TAPS * 16 + 255) / 256), dim3(256), 0, stream>>>(scales, wbank);
    cwt_wmma_kernel<<<dim3(NPIX / 256, 16), dim3(256), 0, stream>>>(x, wbank, out);
}